// TwinGCN_90366111908400
// MI455X (gfx1250) — compile-verified
//
#include <hip/hip_runtime.h>
#include <hip/hip_bf16.h>

typedef __attribute__((ext_vector_type(2))) float v2f;
typedef __attribute__((ext_vector_type(8))) float v8f;

#define NHID 64
#define NCLASS 40
#define NLAYER 4

// ---------------- degree / norm ----------------
__global__ void k_init_deg(float* deg, int N) {
    int i = blockIdx.x * blockDim.x + threadIdx.x;
    if (i < N) deg[i] = 1.0f;  // +1 self loop
}

__global__ void k_deg_edges(float* deg, const long long* __restrict__ dst, int E) {
    int e = blockIdx.x * blockDim.x + threadIdx.x;
    if (e < E) atomicAdd(&deg[(int)dst[e]], 1.0f);
}

__global__ void k_dinv(float* d, int N) {
    int i = blockIdx.x * blockDim.x + threadIdx.x;
    if (i < N) d[i] = rsqrtf(d[i]);   // dinv; self_norm = dinv*dinv
}

// ---------------- fp32 WMMA GEMM: Out[N x 64] = X[N x K] @ W[K x 64] ----------------
// 8 waves/block; wave w -> output tile (row_tile = blkIdx*2 + w/4, col_tile = w%4)
// V_WMMA_F32_16X16X4_F32 layouts (ISA 7.12.2):
//   A 16x4: lanes 0-15 M=0..15 K-off {0,1}; lanes 16-31 same M, K-off {2,3}
//   B 4x16: lanes give N=l%16; VGPR j low lanes K=j, high lanes K=j+2
//   D: VGPR v -> M = v (lanes 0-15) / v+8 (lanes 16-31), N = l%16
__global__ void k_gemm_wmma(const float* __restrict__ X, const float* __restrict__ W,
                            float* __restrict__ Out, int N, int K) {
    int lane = threadIdx.x & 31;
    int wv   = threadIdx.x >> 5;
    int row0 = (blockIdx.x * 2 + (wv >> 2)) * 16;
    int col0 = (wv & 3) * 16;
    int half = lane >> 4;     // 0: lanes 0-15, 1: lanes 16-31
    int lm   = lane & 15;

    int m  = row0 + lm;
    int ms = (m < N) ? m : (N - 1);          // clamp: garbage rows never stored
    const float* xrow = X + (size_t)ms * K;

    v8f acc = {};
    for (int k = 0; k < K; k += 4) {
        int ka = k + half * 2;
        v2f a, b;
        a.x = xrow[ka];
        a.y = xrow[ka + 1];
        b.x = W[(size_t)ka       * NHID + col0 + lm];
        b.y = W[(size_t)(ka + 1) * NHID + col0 + lm];
        acc = __builtin_amdgcn_wmma_f32_16x16x4_f32(
                  false, a, false, b, (short)0, acc, false, false);
    }
#pragma unroll
    for (int v = 0; v < 8; v++) {
        int row = row0 + half * 8 + v;
        if (row < N) Out[(size_t)row * NHID + col0 + lm] = acc[v];
    }
}

// ---------------- edge scatter: agg[dst] += h[src] * dinv[src]*dinv[dst] ----------------
// 64 consecutive threads per edge -> fully coalesced 256B gather + contiguous atomics
__global__ void k_scatter(const float* __restrict__ h, float* __restrict__ agg,
                          const long long* __restrict__ src, const long long* __restrict__ dst,
                          const float* __restrict__ dinv, int E) {
    long long idx = (long long)blockIdx.x * blockDim.x + threadIdx.x;
    int e = (int)(idx >> 6);
    int d = (int)(idx & 63);
    if (e < E) {
        int s = (int)src[e];
        int t = (int)dst[e];
        float w = dinv[s] * dinv[t];
        atomicAdd(&agg[(size_t)t * NHID + d], h[(size_t)s * NHID + d] * w);
    }
}

// ---------------- x_next = relu(agg + h*self_norm + b) ----------------
__global__ void k_finalize(const float* __restrict__ agg, const float* __restrict__ h,
                           const float* __restrict__ dinv, const float* __restrict__ b,
                           float* __restrict__ Hl, int N) {
    long long idx = (long long)blockIdx.x * blockDim.x + threadIdx.x;
    if (idx < (long long)N * NHID) {
        int n = (int)(idx >> 6);
        int d = (int)(idx & 63);
        float di = dinv[n];
        float v = agg[idx] + h[idx] * (di * di) + b[d];
        Hl[idx] = fmaxf(v, 0.0f);
    }
}

// ---------------- layer attention (twin path == main path numerically) ----------------
// one wave per node; lane covers dims {lane, lane+32}
__global__ void k_attn(const float* __restrict__ H, float* __restrict__ hc, int N) {
    int lane = threadIdx.x & 31;
    int node = blockIdx.x * (blockDim.x >> 5) + (threadIdx.x >> 5);
    if (node >= N) return;  // wave-uniform exit
    size_t stride = (size_t)N * NHID;
    const float* p = H + (size_t)node * NHID;

    float s[NLAYER];
#pragma unroll
    for (int l = 0; l < NLAYER; l++) {
        float v0 = p[l * stride + lane];
        float v1 = p[l * stride + lane + 32];
        float part = v0 * v0 + v1 * v1;
#pragma unroll
        for (int off = 16; off > 0; off >>= 1) part += __shfl_xor(part, off, 32);
        s[l] = part;  // TEMPERATURE == 1.0
    }
    float mx = fmaxf(fmaxf(s[0], s[1]), fmaxf(s[2], s[3]));
    float e[NLAYER], tot = 0.0f;
#pragma unroll
    for (int l = 0; l < NLAYER; l++) { e[l] = __expf(s[l] - mx); tot += e[l]; }
    float inv = 1.0f / tot;
    float h0 = 0.0f, h1 = 0.0f;
#pragma unroll
    for (int l = 0; l < NLAYER; l++) {
        float a = e[l] * inv;
        h0 += a * p[l * stride + lane];
        h1 += a * p[l * stride + lane + 32];
    }
    hc[(size_t)node * NHID + lane]      = h0;
    hc[(size_t)node * NHID + lane + 32] = h1;
}

// ---------------- out = hc @ Wout + bout ----------------
__global__ void k_proj(const float* __restrict__ hc, const float* __restrict__ Wout,
                       const float* __restrict__ bout, float* __restrict__ out, int N) {
    long long idx = (long long)blockIdx.x * blockDim.x + threadIdx.x;
    if (idx < (long long)N * NCLASS) {
        int n = (int)(idx / NCLASS);
        int c = (int)(idx % NCLASS);
        const float* hp = hc + (size_t)n * NHID;
        float sacc = bout[c];
#pragma unroll
        for (int d = 0; d < NHID; d++) sacc += hp[d] * Wout[d * NCLASS + c];
        out[idx] = sacc;
    }
}

extern "C" void kernel_launch(void* const* d_in, const int* in_sizes, int n_in,
                              void* d_out, int out_size, void* d_ws, size_t ws_size,
                              hipStream_t stream) {
    const float*     x    = (const float*)d_in[0];
    const float*     W0   = (const float*)d_in[1];
    const float*     b0   = (const float*)d_in[2];
    const float*     Ws   = (const float*)d_in[3];
    const float*     bs   = (const float*)d_in[4];
    const float*     Wout = (const float*)d_in[5];
    const float*     bout = (const float*)d_in[6];
    const long long* ei   = (const long long*)d_in[7];

    int N = in_sizes[0] / 128;   // 50000
    int E = in_sizes[7] / 2;     // 800000
    const long long* src = ei;
    const long long* dst = ei + E;

    float* ws   = (float*)d_ws;
    float* dinv = ws;                                   // N
    float* Hb   = ws + N;                               // NLAYER * N * 64
    float* hbuf = Hb + (size_t)NLAYER * N * NHID;       // N * 64 (also reused as hcomb)
    float* agg  = hbuf + (size_t)N * NHID;              // N * 64

    const int T = 256;

    k_init_deg<<<(N + T - 1) / T, T, 0, stream>>>(dinv, N);
    k_deg_edges<<<(E + T - 1) / T, T, 0, stream>>>(dinv, dst, E);
    k_dinv<<<(N + T - 1) / T, T, 0, stream>>>(dinv, N);

    for (int l = 0; l < NLAYER; l++) {
        const float* xin = (l == 0) ? x : Hb + (size_t)(l - 1) * N * NHID;
        int K            = (l == 0) ? 128 : NHID;
        const float* W   = (l == 0) ? W0 : Ws + (size_t)(l - 1) * NHID * NHID;
        const float* b   = (l == 0) ? b0 : bs + (size_t)(l - 1) * NHID;

        k_gemm_wmma<<<(N + 31) / 32, 256, 0, stream>>>(xin, W, hbuf, N, K);
        hipMemsetAsync(agg, 0, (size_t)N * NHID * sizeof(float), stream);
        long long te = (long long)E * NHID;
        k_scatter<<<(unsigned)((te + T - 1) / T), T, 0, stream>>>(hbuf, agg, src, dst, dinv, E);
        long long tn = (long long)N * NHID;
        k_finalize<<<(unsigned)((tn + T - 1) / T), T, 0, stream>>>(
            agg, hbuf, dinv, b, Hb + (size_t)l * N * NHID, N);
    }

    k_attn<<<(N + 7) / 8, 256, 0, stream>>>(Hb, hbuf, N);
    long long to = (long long)N * NCLASS;
    k_proj<<<(unsigned)((to + T - 1) / T), T, 0, stream>>>(hbuf, Wout, bout, (float*)d_out, N);
}